// NCPLiquidCell_35175782154587
// MI455X (gfx1250) — compile-verified
//
#include <hip/hip_runtime.h>
#include <hip/hip_bf16.h>
#include <math.h>

// NCP Liquid Cell for MI455X (gfx1250, wave32).
// 5 masked fp32 GEMMs fused (shared xc input, shared mask), computed with
// V_WMMA_F32_16X16X4_F32, followed by fused tanh/sigmoid liquid-cell epilogue.
//
// Sizes: B=2048, IN_F=512, N_HID=2048, HEAD=2560.
// Memory-bound (~180MB @ 23.3TB/s); weights+mask fit in 192MB L2.

typedef __attribute__((ext_vector_type(2))) float v2f;
typedef __attribute__((ext_vector_type(4))) float v4f;
typedef __attribute__((ext_vector_type(8))) float v8f;

#define BDIM   2048
#define IN_F   512
#define N_HID  2048
#define HEADK  2560

#define BM 64
#define BN 32
#define BK 64
#define LDA 68   // padded f32 stride: no LDS bank conflicts

__global__ __launch_bounds__(256, 2)
void ncp_liquid_cell_wmma_f32(const float* __restrict__ x,
                              const float* __restrict__ hidden,
                              const float* __restrict__ mask,
                              const float* __restrict__ Wg,
                              const float* __restrict__ Wh,
                              const float* __restrict__ Wfg,
                              const float* __restrict__ Wfh,
                              const float* __restrict__ Wp,
                              const float* __restrict__ bg,
                              const float* __restrict__ bh,
                              const float* __restrict__ bfg,
                              const float* __restrict__ bfh,
                              const float* __restrict__ bp,
                              float* __restrict__ out)
{
    __shared__ float sX[BM * LDA];          // xc tile  [64 x 64]
    __shared__ float sW[5][BN * LDA];       // masked weight tiles [32 x 64] x5

    const float* const Ws[5] = {Wg, Wh, Wfg, Wfh, Wp};

    const int t    = threadIdx.x;
    const int m0   = blockIdx.x * BM;       // batch-row tile
    const int n0   = blockIdx.y * BN;       // hidden-col tile
    const int wave = t >> 5;
    const int lane = t & 31;
    const int wm   = wave & 3;              // 4 row subtiles of 16
    const int wn   = wave >> 2;             // 2 col subtiles of 16
    const int msub = wm * 16;
    const int nsub = wn * 16;
    const int row16 = lane & 15;
    const int half  = lane >> 4;            // 0: K=0,1  1: K=2,3 (A/B frag layout)
    const int koff  = half * 2;

    v8f acc[5];
#pragma unroll
    for (int m = 0; m < 5; ++m)
#pragma unroll
        for (int j = 0; j < 8; ++j) acc[m][j] = 0.0f;

    for (int k0 = 0; k0 < HEADK; k0 += BK) {
        __syncthreads();

        // ---- Stage xc tile (concat of x | hidden along K) into LDS ----
        // k0 is a multiple of 64 and the split is at 512, so each tile lies
        // entirely in x or entirely in hidden: block-uniform select, no lane
        // divergence around the WMMAs.
        {
            const bool inx       = (k0 < IN_F);
            const float* src     = inx ? x : hidden;
            const int    stride  = inx ? IN_F : N_HID;
            const int    kbase   = inx ? k0 : (k0 - IN_F);
#pragma unroll
            for (int i = 0; i < 4; ++i) {
                const int id = t + 256 * i;          // 1024 float4 total
                const int r  = id >> 4;
                const int c4 = (id & 15) * 4;
                v4f v = *reinterpret_cast<const v4f*>(
                    src + (size_t)(m0 + r) * stride + kbase + c4);
                *reinterpret_cast<v4f*>(&sX[r * LDA + c4]) = v;
            }
            // prefetch next K-tile of xc (global_prefetch_b8)
            if (k0 + BK < HEADK) {
                const bool nx      = (k0 + BK < IN_F);
                const float* psrc  = nx ? x : hidden;
                const int   pstr   = nx ? IN_F : N_HID;
                const int   pk     = nx ? (k0 + BK) : (k0 + BK - IN_F);
                const int   r      = t >> 4;
                const int   c4     = (t & 15) * 4;
                __builtin_prefetch(psrc + (size_t)(m0 + r) * pstr + pk + c4, 0, 0);
            }
        }

        // ---- Stage 5 weight tiles, applying the shared mask on the fly ----
#pragma unroll
        for (int i = 0; i < 2; ++i) {
            const int id = t + 256 * i;              // 512 float4 per matrix
            const int r  = id >> 4;
            const int c4 = (id & 15) * 4;
            const size_t goff = (size_t)(n0 + r) * HEADK + k0 + c4;
            v4f mv = *reinterpret_cast<const v4f*>(mask + goff);
#pragma unroll
            for (int mtx = 0; mtx < 5; ++mtx) {
                v4f wv = *reinterpret_cast<const v4f*>(Ws[mtx] + goff);
                wv *= mv;
                *reinterpret_cast<v4f*>(&sW[mtx][r * LDA + c4]) = wv;
            }
            if (k0 + BK < HEADK)
                __builtin_prefetch(mask + goff + BK, 0, 0);
        }

        __syncthreads();

        // ---- Compute: 16 K-steps x 5 WMMAs (v_wmma_f32_16x16x4_f32) ----
#pragma unroll
        for (int kk = 0; kk < BK; kk += 4) {
            // A fragment: 16x4 f32. lane<16 -> K=kk,kk+1 ; lane>=16 -> K=kk+2,kk+3
            v2f a = *reinterpret_cast<const v2f*>(
                &sX[(msub + row16) * LDA + kk + koff]);
#pragma unroll
            for (int mtx = 0; mtx < 5; ++mtx) {
                // B fragment: 4x16 f32, B[k][n] = (W*mask)[n][k]
                v2f b = *reinterpret_cast<const v2f*>(
                    &sW[mtx][(nsub + row16) * LDA + kk + koff]);
                acc[mtx] = __builtin_amdgcn_wmma_f32_16x16x4_f32(
                    false, a, false, b, (short)0, acc[mtx], false, false);
            }
        }
    }

    // ---- Fused liquid-cell epilogue ----
    // C/D layout: VGPR i, lanes 0-15: M=i, N=lane ; lanes 16-31: M=i+8, N=lane-16
    const int gcol = n0 + nsub + row16;
    const float bgv  = bg[gcol];
    const float bhv  = bh[gcol];
    const float bfgv = bfg[gcol];
    const float bfhv = bfh[gcol];
    const float bpv  = bp[gcol];

    const size_t NH_OUT = (size_t)BDIM * N_HID;   // offset of new_hidden in d_out

#pragma unroll
    for (int i = 0; i < 8; ++i) {
        const int grow = m0 + msub + i + half * 8;
        const float g_out  = acc[0][i] + bgv;
        const float h_out  = acc[1][i] + bhv;
        const float f_out  = (acc[2][i] + bfgv) + (acc[3][i] + bfhv);
        const float p_out  = acc[4][i] + bpv;
        const float g    = tanhf(g_out);
        const float h    = tanhf(h_out);
        const float gate = 1.0f / (1.0f + __expf(-f_out));
        const float nh   = g * (1.0f - gate) + gate * h;
        const size_t idx = (size_t)grow * N_HID + gcol;
        out[idx]          = p_out + nh;   // y_pred
        out[NH_OUT + idx] = nh;           // new_hidden
    }
}

extern "C" void kernel_launch(void* const* d_in, const int* in_sizes, int n_in,
                              void* d_out, int out_size, void* d_ws, size_t ws_size,
                              hipStream_t stream) {
    (void)in_sizes; (void)n_in; (void)out_size; (void)d_ws; (void)ws_size;
    const float* x      = (const float*)d_in[0];
    const float* hidden = (const float*)d_in[1];
    const float* mask   = (const float*)d_in[2];
    const float* Wg     = (const float*)d_in[3];
    const float* bg     = (const float*)d_in[4];
    const float* Wh     = (const float*)d_in[5];
    const float* bh     = (const float*)d_in[6];
    const float* Wfg    = (const float*)d_in[7];
    const float* bfg    = (const float*)d_in[8];
    const float* Wfh    = (const float*)d_in[9];
    const float* bfh    = (const float*)d_in[10];
    const float* Wp     = (const float*)d_in[11];
    const float* bp     = (const float*)d_in[12];
    float* out          = (float*)d_out;

    dim3 grid(BDIM / BM, N_HID / BN);   // 32 x 64 blocks
    dim3 block(256);                    // 8 wave32 per block
    ncp_liquid_cell_wmma_f32<<<grid, block, 0, stream>>>(
        x, hidden, mask, Wg, Wh, Wfg, Wfh, Wp, bg, bh, bfg, bfh, bp, out);
}